// WDerivativeDILATEloss_35476429865786
// MI455X (gfx1250) — compile-verified
//
#include <hip/hip_runtime.h>

// Problem constants (from the reference)
#define Mlen 319          // N-1
#define NPTS 320          // N, also blockDim for DP kernel (10 wave32s)
#define BATCH 128

static constexpr float ALPHA = 1.0f;
static constexpr float BETA  = 0.1f;
static constexpr float INFV  = 1e10f;
// softmin with gamma=0.01 via exp2/log2 (v_exp_f32 / v_log_f32 are base-2):
//   K_EXP = (1/gamma)*log2(e),  K_LOG = gamma*ln(2)
static constexpr float K_EXP = 144.2695040888963f;
static constexpr float K_LOG = 0.006931471805599453f;

// raw ISA transcendentals (v_exp_f32 / v_log_f32 are base-2 on AMDGPU)
__device__ __forceinline__ float fast_exp2(float x) { return __builtin_amdgcn_exp2f(x); }
__device__ __forceinline__ float fast_log2(float x) { return __builtin_amdgcn_logf(x); }

typedef __attribute__((ext_vector_type(2))) float v2f;
typedef __attribute__((ext_vector_type(8))) float v8f;

// -----------------------------------------------------------------------------
// Kernel 1: per-batch soft-DTW forward (checkpoint R to ws) + backward gradient
// contraction with Omega. One block per batch, one thread per DP row, anti-
// diagonals in LDS, one workgroup barrier per wavefront step.
// -----------------------------------------------------------------------------
__global__ __launch_bounds__(NPTS) void dilate_dp_kernel(
    const float* __restrict__ inp, const float* __restrict__ tgt,
    float* __restrict__ Rws, float* __restrict__ ptemp)
{
  const int b   = blockIdx.x;
  const int tid = threadIdx.x;          // row index i = tid (0..M)

  __shared__ float di_s[Mlen];
  __shared__ float dt_s[Mlen];
  __shared__ float bufA[NPTS + 2], bufB[NPTS + 2], bufC[NPTS + 2]; // fwd R / bwd E diags
  __shared__ float rA[NPTS + 2], rB[NPTS + 2], rC[NPTS + 2];       // bwd R diags
  __shared__ float red[NPTS];

  if (tid < Mlen) {
    di_s[tid] = inp[b * NPTS + tid + 1] - inp[b * NPTS + tid];
    dt_s[tid] = tgt[b * NPTS + tid + 1] - tgt[b * NPTS + tid];
  }
  bufA[tid] = (tid == 0) ? 0.0f : INFV;   // diagonal k=0
  bufB[tid] = INFV;                        // diagonal k=1
  __syncthreads();

  const int   i     = tid;
  const bool  rowOK = (i >= 1);
  const float dti   = rowOK ? dt_s[i - 1] : 0.0f;

  // ---------------- forward pass ----------------
  float* prev2 = bufA;
  float* prev  = bufB;
  float* nxt   = bufC;

  for (int k = 2; k <= 2 * Mlen; ++k) {
    const int j = k - i;
    float v = INFV;
    if (rowOK && j >= 1 && j <= Mlen) {
      float a  = prev2[i - 1];                  // R[i-1,j-1]
      float bb = prev [i - 1];                  // R[i-1,j]
      float c  = prev [i];                      // R[i,j-1]
      float m  = fminf(a, fminf(bb, c));
      float z  = fast_exp2((m - a) * K_EXP) + fast_exp2((m - bb) * K_EXP)
               + fast_exp2((m - c) * K_EXP);
      float s  = m - K_LOG * fast_log2(z);      // softmin
      float d  = dti - di_s[j - 1];
      v = fmaf(d, d, s);
      Rws[((size_t)b * Mlen + (i - 1)) * Mlen + (j - 1)] = v;
    }
    nxt[i] = v;
    __syncthreads();
    float* t = prev2; prev2 = prev; prev = nxt; nxt = t;
  }

  // ---------------- backward pass (E recursion) ----------------
  bufA[tid] = 0.0f; bufB[tid] = 0.0f; bufC[tid] = 0.0f;   // E diagonals
  rA[tid] = INFV;   rB[tid] = INFV;   rC[tid] = INFV;     // R diagonals (guarded anyway)
  __syncthreads();

  float* eC   = bufA; float* eN = bufB; float* eN2 = bufC; // diag k, k+1, k+2
  float* rCur = rA;   float* rN = rB;   float* rN2 = rC;

  const float dtip = (i < Mlen) ? dt_s[i] : 0.0f;          // dt for row i+1
  float tempAcc = 0.0f;

  // software-pipelined R load (L2-resident: 52MB R fits in 192MB L2)
  float rload = 0.0f;
  {
    const int j0 = 2 * Mlen - i;
    if (rowOK && j0 >= 1 && j0 <= Mlen)
      rload = Rws[((size_t)b * Mlen + (i - 1)) * Mlen + (j0 - 1)];
  }

  for (int k = 2 * Mlen; k >= 2; --k) {
    const int  j     = k - i;
    const bool valid = rowOK && (j >= 1) && (j <= Mlen);
    const float rij  = rload;
    {   // prefetch the R value for the next diagonal
      const int jn = (k - 1) - i;
      if (rowOK && jn >= 1 && jn <= Mlen)
        rload = Rws[((size_t)b * Mlen + (i - 1)) * Mlen + (jn - 1)];
    }

    float e = 0.0f;
    if (valid) {
      if (i == Mlen && j == Mlen) {
        e = 1.0f;                                  // E[M,M] seed
      } else {
        float acc = 0.0f;
        if (i < Mlen) {                            // a: via (i+1, j)
          float d = dtip - di_s[j - 1];
          acc = fmaf(fast_exp2((rN[i + 1] - rij - d * d) * K_EXP), eN[i + 1], acc);
        }
        if (j < Mlen) {                            // b: via (i, j+1)
          float d = dti - di_s[j];
          acc = fmaf(fast_exp2((rN[i] - rij - d * d) * K_EXP), eN[i], acc);
        }
        if (i < Mlen && j < Mlen) {                // c: via (i+1, j+1)
          float d = dtip - di_s[j];
          acc = fmaf(fast_exp2((rN2[i + 1] - rij - d * d) * K_EXP), eN2[i + 1], acc);
        }
        e = acc;
      }
      const float w = (float)(i - j);
      tempAcc = fmaf(e, w * w, tempAcc);           // sum E * Omega on the fly
    }
    eC[i]   = e;
    rCur[i] = rij;
    __syncthreads();
    float* t1 = eN2; eN2 = eN; eN = eC;   eC   = t1;
    float* t2 = rN2; rN2 = rN; rN = rCur; rCur = t2;
  }

  // block reduction of tempAcc (fixed order -> deterministic)
  red[tid] = tempAcc;
  __syncthreads();
  if (tid < 64)
    red[tid] = red[tid] + red[tid + 64] + red[tid + 128] + red[tid + 192] + red[tid + 256];
  __syncthreads();
  if (tid < 32) red[tid] += red[tid + 32];
  __syncthreads();
  if (tid < 32) {
    float s = red[tid];
    for (int off = 16; off; off >>= 1) s += __shfl_down(s, off, 32);
    if (tid == 0) ptemp[b] = s;
  }
}

// -----------------------------------------------------------------------------
// Kernel 2: combine per-batch partials into the scalar loss. The 128-way batch
// reduction runs through the CDNA5 matrix pipe: two v_wmma_f32_16x16x4_f32
// against an all-ones B compute row-sums, C-chained. One full wave32 (EXEC all
// ones, as WMMA requires).
// -----------------------------------------------------------------------------
__global__ __launch_bounds__(32) void dilate_reduce_kernel(
    const float* __restrict__ Rws, const float* __restrict__ ptemp,
    float* __restrict__ out)
{
  const int lane = threadIdx.x;               // 0..31
  const float cs = ALPHA / (float)BATCH;
  const float ct = BETA / ((float)BATCH * (float)Mlen * (float)Mlen);

  // A-matrix layout for 32-bit A 16x4: lane L (0-15): v0=A[L,0], v1=A[L,1];
  // lane 16+L: v0=A[L,2], v1=A[L,3].  A[m,k] = partial[4m+k] (+64 for 2nd WMMA).
  const int m  = lane & 15;
  const int kh = (lane >> 4) << 1;            // 0 or 2

  const int i0 = 4 * m + kh;
  const int i1 = i0 + 1;

  #define COMBINE(idx) \
    (cs * Rws[((size_t)(idx) * Mlen + (Mlen - 1)) * Mlen + (Mlen - 1)] + ct * ptemp[(idx)])

  v2f a0, a1, bones;
  a0.x = COMBINE(i0);      a0.y = COMBINE(i1);
  a1.x = COMBINE(i0 + 64); a1.y = COMBINE(i1 + 64);
  #undef COMBINE
  bones.x = 1.0f; bones.y = 1.0f;             // all-ones B => row sums

  v8f c = {};
  c = __builtin_amdgcn_wmma_f32_16x16x4_f32(false, a0, false, bones, (short)0, c,
                                            false, false);
  c = __builtin_amdgcn_wmma_f32_16x16x4_f32(false, a1, false, bones, (short)0, c,
                                            false, false);

  // D[m,n] identical across n; lane holds column n=lane&15 for 8 rows.
  float s = c[0] + c[1] + c[2] + c[3] + c[4] + c[5] + c[6] + c[7];
  float tot = __shfl(s, 0, 32) + __shfl(s, 16, 32);  // rows 0-7 + rows 8-15 at col 0
  if (lane == 0) out[0] = tot;
}

// -----------------------------------------------------------------------------
extern "C" void kernel_launch(void* const* d_in, const int* in_sizes, int n_in,
                              void* d_out, int out_size, void* d_ws, size_t ws_size,
                              hipStream_t stream) {
  const float* inp = (const float*)d_in[0];   // (B, N, 1) f32
  const float* tgt = (const float*)d_in[1];   // (B, N, 1) f32
  float* Rws   = (float*)d_ws;                             // B*M*M floats (~52MB)
  float* ptemp = Rws + (size_t)BATCH * Mlen * Mlen;        // B floats
  float* out   = (float*)d_out;

  dilate_dp_kernel<<<BATCH, NPTS, 0, stream>>>(inp, tgt, Rws, ptemp);
  dilate_reduce_kernel<<<1, 32, 0, stream>>>(Rws, ptemp, out);
}